// EfficientAttention_27178553049199
// MI455X (gfx1250) — compile-verified
//
#include <hip/hip_runtime.h>
#include <math.h>

// Linear attention, N=262144, d_k=d_v=64.  Memory-bound (>=256MB traffic,
// ~11-15us floor at 23.3TB/s).  f32 WMMA (16x16x4) for both matmuls,
// async global->LDS (b128) for K/V tile staging in the context pass.

#define NROWS 262144
#define DIM   64
#define KEXP  0.18033688011112042f   // (1/8) * log2(e)

#define CTX_BLOCKS 512
#define CTX_WAVES  4                 // 512 x 4 = 2048 waves streaming K/V

#define USE_ASYNC_LDS 1

typedef float v2f __attribute__((ext_vector_type(2)));
typedef float v8f __attribute__((ext_vector_type(8)));

static __device__ __forceinline__ v8f wmma_f32_16x16x4(v2f a, v2f b, v8f c) {
  // 8 args: (neg_a, A, neg_b, B, c_mod, C, reuse_a, reuse_b)
  return __builtin_amdgcn_wmma_f32_16x16x4_f32(false, a, false, b, (short)0, c,
                                               false, false);
}

// Low 32 bits of a generic pointer into LDS == wave-relative LDS byte offset.
static __device__ __forceinline__ unsigned ldsOffsetOf(const void* p) {
  return (unsigned)(size_t)p;
}

static __device__ __forceinline__ void asyncLoadLdsB128(unsigned ldsOff,
                                                        const float* gaddr) {
#if USE_ASYNC_LDS
  asm volatile("global_load_async_to_lds_b128 %0, %1, off"
               :: "v"(ldsOff), "v"(gaddr) : "memory");
#endif
}

static __device__ __forceinline__ void waitAsync0() {
#if USE_ASYNC_LDS
  asm volatile("s_wait_asynccnt 0x0" ::: "memory");
#endif
}

// ---------- Pass 1: per-block column maxes of Q (b128 loads) ----------
// element (i*4+k) column = (tid*4+k) & 63 since strides are multiples of 64.
__global__ __launch_bounds__(256) void kColMaxPartial(const float* __restrict__ Q,
                                                      float* __restrict__ pmax) {
  __shared__ float sm[64][16];
  const int tid = threadIdx.x, bid = blockIdx.x;
  const int epb4 = (NROWS / 256) * DIM / 4;         // 16384 float4 / block
  const float4* q4 = (const float4*)(Q + (size_t)bid * epb4 * 4);
  float4 mx = {-__builtin_inff(), -__builtin_inff(),
               -__builtin_inff(), -__builtin_inff()};
  for (int i = tid; i < epb4; i += 256) {
    float4 v = q4[i];
    mx.x = fmaxf(mx.x, v.x); mx.y = fmaxf(mx.y, v.y);
    mx.z = fmaxf(mx.z, v.z); mx.w = fmaxf(mx.w, v.w);
  }
  const int c0 = (tid * 4) & 63;                    // 4 columns owned
  const int mem = tid >> 4;                         // 16 members per column set
  sm[c0 + 0][mem] = mx.x; sm[c0 + 1][mem] = mx.y;
  sm[c0 + 2][mem] = mx.z; sm[c0 + 3][mem] = mx.w;
  __syncthreads();
  if (tid < 64) {
    float v = -__builtin_inff();
    for (int j = 0; j < 16; ++j) v = fmaxf(v, sm[tid][j]);
    pmax[bid * 64 + tid] = v;
  }
}

__global__ __launch_bounds__(64) void kReduceMax(const float* __restrict__ pmax,
                                                 float* __restrict__ m) {
  const int c = threadIdx.x;
  float mx = -__builtin_inff();
  for (int b = 0; b < 256; ++b) mx = fmaxf(mx, pmax[b * 64 + c]);
  m[c] = mx;
}

// ---------- Pass 2: per-block column sum of exp (b128 loads) ----------
__global__ __launch_bounds__(256) void kSumExpPartial(const float* __restrict__ Q,
                                                      const float* __restrict__ m,
                                                      float* __restrict__ psum) {
  __shared__ float sm[64][16];
  const int tid = threadIdx.x, bid = blockIdx.x;
  const int epb4 = (NROWS / 256) * DIM / 4;
  const float4* q4 = (const float4*)(Q + (size_t)bid * epb4 * 4);
  const int c0 = (tid * 4) & 63;
  const float m0 = m[c0], m1 = m[c0 + 1], m2 = m[c0 + 2], m3 = m[c0 + 3];
  float4 s = {0.f, 0.f, 0.f, 0.f};
  for (int i = tid; i < epb4; i += 256) {
    float4 v = q4[i];
    s.x += exp2f((v.x - m0) * KEXP); s.y += exp2f((v.y - m1) * KEXP);
    s.z += exp2f((v.z - m2) * KEXP); s.w += exp2f((v.w - m3) * KEXP);
  }
  const int mem = tid >> 4;
  sm[c0 + 0][mem] = s.x; sm[c0 + 1][mem] = s.y;
  sm[c0 + 2][mem] = s.z; sm[c0 + 3][mem] = s.w;
  __syncthreads();
  if (tid < 64) {
    float t = 0.f;
    for (int j = 0; j < 16; ++j) t += sm[tid][j];
    psum[bid * 64 + tid] = t;
  }
}

__global__ __launch_bounds__(64) void kReduceSum(const float* __restrict__ psum,
                                                 float* __restrict__ s) {
  const int c = threadIdx.x;
  float t = 0.0f;
  for (int b = 0; b < 256; ++b) t += psum[b * 64 + c];
  s[c] = t;
}

// ---------- Pass 3: context partials, ctx[c,v] += sum_i kn[i,c]*V[i,v] ----------
// 512 blocks x 4 waves; each wave owns private 16x64 K/V tiles (async-filled),
// accumulates the full 64x64 in 16 f32 WMMA tiles, then a fixed-order
// (deterministic) cross-wave merge into block LDS.
__global__ __launch_bounds__(CTX_WAVES * 32) void kCtxPartial(
    const float* __restrict__ K, const float* __restrict__ V,
    float* __restrict__ pctx) {
  __shared__ float ldsK[CTX_WAVES][16 * 64];
  __shared__ float ldsV[CTX_WAVES][16 * 64];
  __shared__ float ldsAcc[4096];
  const int tid  = threadIdx.x;
  const int wave = tid >> 5;
  const int lane = tid & 31;
  const int bid  = blockIdx.x;
  const int half = lane >> 4;
  const int mloc = lane & 15;

  for (int e = tid; e < 4096; e += CTX_WAVES * 32) ldsAcc[e] = 0.0f;

  v8f acc[4][4];
#pragma unroll
  for (int t = 0; t < 4; ++t)
#pragma unroll
    for (int u = 0; u < 4; ++u)
      acc[t][u] = (v8f){0.f, 0.f, 0.f, 0.f, 0.f, 0.f, 0.f, 0.f};

  const unsigned kOff = ldsOffsetOf(&ldsK[wave][0]);
  const unsigned vOff = ldsOffsetOf(&ldsV[wave][0]);
  const int rowsPerBlock = NROWS / CTX_BLOCKS;            // 512
  const int rows0 = bid * rowsPerBlock + wave * (rowsPerBlock / CTX_WAVES);
  __syncthreads();                                        // ldsAcc zeroed

  for (int chunk = 0; chunk < (rowsPerBlock / CTX_WAVES) / 16; ++chunk) {
    const int rowbase = rows0 + chunk * 16;
    const float* kp = K + (size_t)rowbase * 64;
    const float* vp = V + (size_t)rowbase * 64;
#if USE_ASYNC_LDS
#pragma unroll
    for (int j = 0; j < 8; ++j) {                         // 16B per lane per op
      const int f4 = lane + j * 32;
      asyncLoadLdsB128(kOff + f4 * 16, kp + f4 * 4);
      asyncLoadLdsB128(vOff + f4 * 16, vp + f4 * 4);
    }
    waitAsync0();
#else
#pragma unroll
    for (int j = 0; j < 8; ++j) {
      const int f4 = lane + j * 32;
      ((float4*)ldsK[wave])[f4] = ((const float4*)kp)[f4];
      ((float4*)ldsV[wave])[f4] = ((const float4*)vp)[f4];
    }
#endif
    // per-row softmax of the K tile (lanes 0-15 own one row each; per-wave
    // LDS ordering: DS ops from one wave complete in order)
    if (lane < 16) {
      float* row = &ldsK[wave][lane * 64];
      float mx = -__builtin_inff();
      for (int c = 0; c < 64; ++c) mx = fmaxf(mx, row[c]);
      float sum = 0.0f;
      for (int c = 0; c < 64; ++c) {
        float e = exp2f((row[c] - mx) * KEXP); row[c] = e; sum += e;
      }
      float inv = 1.0f / sum;
      for (int c = 0; c < 64; ++c) row[c] *= inv;
    }
    __syncthreads();
#pragma unroll
    for (int ks = 0; ks < 4; ++ks) {
      const int r0 = ks * 4 + 2 * half;   // this lane-half's first K row
      v2f a[4], b[4];
#pragma unroll
      for (int t = 0; t < 4; ++t) {       // A = kn^T : A[c, i] = kn[i, c]
        int c = t * 16 + mloc;
        a[t].x = ldsK[wave][r0 * 64 + c];
        a[t].y = ldsK[wave][(r0 + 1) * 64 + c];
      }
#pragma unroll
      for (int u = 0; u < 4; ++u) {       // B : B[i, v] = V[i, v]
        int v = u * 16 + mloc;
        b[u].x = ldsV[wave][r0 * 64 + v];
        b[u].y = ldsV[wave][(r0 + 1) * 64 + v];
      }
#pragma unroll
      for (int t = 0; t < 4; ++t)
#pragma unroll
        for (int u = 0; u < 4; ++u)
          acc[t][u] = wmma_f32_16x16x4(a[t], b[u], acc[t][u]);
    }
    __syncthreads();
  }

  // deterministic cross-wave merge (fixed wave order; disjoint (c,v) per lane)
  for (int w = 0; w < CTX_WAVES; ++w) {
    if (wave == w) {
#pragma unroll
      for (int t = 0; t < 4; ++t)
#pragma unroll
        for (int u = 0; u < 4; ++u)
#pragma unroll
          for (int r = 0; r < 8; ++r) {
            int c = t * 16 + r + 8 * half;    // D: VGPR r -> M=r / M=r+8
            int v = u * 16 + mloc;
            ldsAcc[c * 64 + v] += acc[t][u][r];
          }
    }
    __syncthreads();
  }
  float* dst = pctx + (size_t)bid * 4096;
  for (int e = tid; e < 4096; e += CTX_WAVES * 32) dst[e] = ldsAcc[e];
}

// ---------- Reduce context partials (fixed order) and fold in 1/s ----------
__global__ __launch_bounds__(256) void kReduceCtx(const float* __restrict__ pctx,
                                                  const float* __restrict__ s,
                                                  float* __restrict__ ctxp) {
  const int e = blockIdx.x * 256 + threadIdx.x;   // 16 blocks x 256 = 4096
  float acc = 0.0f;
  for (int b = 0; b < CTX_BLOCKS; ++b) acc += pctx[(size_t)b * 4096 + e];
  ctxp[e] = acc / s[e >> 6];
}

// ---------- Pass 4: out = E @ ctx'  (E = exp2((Q - m)*KEXP)) ----------
__global__ __launch_bounds__(256) void kOut(const float* __restrict__ Q,
                                            const float* __restrict__ m,
                                            const float* __restrict__ ctxp,
                                            float* __restrict__ out) {
  __shared__ float ldsCtx[4096];
  __shared__ float ldsM[64];
  __shared__ float ldsE[8][1024];
  const int tid  = threadIdx.x;
  const int wave = tid >> 5;
  const int lane = tid & 31;
  const int half = lane >> 4;
  const int mloc = lane & 15;

  for (int e = tid; e < 1024; e += 256)
    ((float4*)ldsCtx)[e] = ((const float4*)ctxp)[e];
  if (tid < 64) ldsM[tid] = m[tid];
  __syncthreads();

  const int rowbase = (blockIdx.x * 8 + wave) * 16;   // 16 rows per wave
  const float4* q4 = (const float4*)(Q + (size_t)rowbase * 64);
  const int c0w = (lane * 4) & 63;                    // cols of this lane's f4
#pragma unroll
  for (int j = 0; j < 8; ++j) {
    int i4 = lane + j * 32;
    float4 v = q4[i4];
    float4 e;
    e.x = exp2f((v.x - ldsM[c0w + 0]) * KEXP);
    e.y = exp2f((v.y - ldsM[c0w + 1]) * KEXP);
    e.z = exp2f((v.z - ldsM[c0w + 2]) * KEXP);
    e.w = exp2f((v.w - ldsM[c0w + 3]) * KEXP);
    ((float4*)ldsE[wave])[i4] = e;
  }
  __syncthreads();

  v8f acc[4];
#pragma unroll
  for (int u = 0; u < 4; ++u)
    acc[u] = (v8f){0.f, 0.f, 0.f, 0.f, 0.f, 0.f, 0.f, 0.f};

#pragma unroll
  for (int ks = 0; ks < 16; ++ks) {
    const int c0 = ks * 4 + 2 * half;
    v2f a;                                  // A[row, c] = E[row, c]
    a.x = ldsE[wave][mloc * 64 + c0];
    a.y = ldsE[wave][mloc * 64 + c0 + 1];
#pragma unroll
    for (int u = 0; u < 4; ++u) {           // B[c, v] = ctx'[c, v]
      v2f b;
      b.x = ldsCtx[c0 * 64 + u * 16 + mloc];
      b.y = ldsCtx[(c0 + 1) * 64 + u * 16 + mloc];
      acc[u] = wmma_f32_16x16x4(a, b, acc[u]);
    }
  }
  float* o = out + (size_t)rowbase * 64;
#pragma unroll
  for (int u = 0; u < 4; ++u)
#pragma unroll
    for (int r = 0; r < 8; ++r) {
      int row = r + 8 * half;
      o[row * 64 + u * 16 + mloc] = acc[u][r];
    }
}

extern "C" void kernel_launch(void* const* d_in, const int* in_sizes, int n_in,
                              void* d_out, int out_size, void* d_ws, size_t ws_size,
                              hipStream_t stream) {
  (void)in_sizes; (void)n_in; (void)out_size; (void)ws_size;
  const float* Q = (const float*)d_in[0];
  const float* K = (const float*)d_in[1];
  const float* V = (const float*)d_in[2];
  float* out = (float*)d_out;

  // ws (floats): m[64] | s[64] | ctxp[4096] | pmax[16384] | psum[16384]
  //              | pctx[512*4096]   (~8.5 MB total)
  float* w    = (float*)d_ws;
  float* m    = w;
  float* s    = w + 64;
  float* ctxp = w + 128;
  float* pmax = w + 128 + 4096;
  float* psum = pmax + 256 * 64;
  float* pctx = psum + 256 * 64;

  kColMaxPartial<<<256, 256, 0, stream>>>(Q, pmax);
  kReduceMax    <<<1,    64, 0, stream>>>(pmax, m);
  kSumExpPartial<<<256, 256, 0, stream>>>(Q, m, psum);
  kReduceSum    <<<1,    64, 0, stream>>>(psum, s);
  kCtxPartial   <<<CTX_BLOCKS, CTX_WAVES * 32, 0, stream>>>(K, V, pctx);
  kReduceCtx    <<<16,  256, 0, stream>>>(pctx, s, ctxp);
  kOut          <<<2048, 256, 0, stream>>>(Q, m, ctxp, out);
}